// Encoder_120259084831
// MI455X (gfx1250) — compile-verified
//
#include <hip/hip_runtime.h>
#include <cstdint>

typedef __attribute__((ext_vector_type(2))) float v2f;
typedef __attribute__((ext_vector_type(8))) float v8f;

#define D 64
#define WPAD 68  // LDS row pad: bank = (4n + k) mod 64 -> conflict-free, 8B-aligned rows

#define WMMA4(a, b, c) \
  __builtin_amdgcn_wmma_f32_16x16x4_f32(false, (a), false, (b), (short)0, (c), false, false)

// Stage a 64x64 row-major weight matrix into LDS *transposed*: wl[n][k] = W[k][n].
// 256 threads, coalesced float4 global reads.
__device__ __forceinline__ void stage_wT(const float* __restrict__ W, float (*wl)[WPAD],
                                         int tid) {
  for (int idx = tid; idx < 64 * 16; idx += 256) {
    const int k = idx >> 4;
    const int n4 = (idx & 15) << 2;
    const float4 v = *reinterpret_cast<const float4*>(W + k * D + n4);
    wl[n4 + 0][k] = v.x;
    wl[n4 + 1][k] = v.y;
    wl[n4 + 2][k] = v.z;
    wl[n4 + 3][k] = v.w;
  }
}

// h0 = log(x + 1); zero deg
__global__ void k_init(const float* __restrict__ x, float* __restrict__ h,
                       float* __restrict__ deg, int n) {
  long i = (long)blockIdx.x * blockDim.x + threadIdx.x;
  long tot = (long)n * D;
  if (i < n) deg[i] = 0.0f;
  if (i < tot) h[i] = __logf(x[i] + 1.0f);
}

// deg[dst] += 1
__global__ void k_deg(const int* __restrict__ dst, float* __restrict__ deg, int e) {
  long i = (long)blockIdx.x * blockDim.x + threadIdx.x;
  if (i < e) atomicAdd(&deg[dst[i]], 1.0f);
}

__global__ void k_zero4(float4* __restrict__ p, long n4) {
  long i = (long)blockIdx.x * blockDim.x + threadIdx.x;
  if (i < n4) p[i] = make_float4(0.f, 0.f, 0.f, 0.f);
}

// neigh[dst] += h[src]; 16 lanes per edge, float4 per lane (coalesced 256B/edge)
__global__ void k_scatter(const float* __restrict__ h, const int* __restrict__ src,
                          const int* __restrict__ dst, float* __restrict__ neigh, int e) {
  long gid = (long)blockIdx.x * blockDim.x + threadIdx.x;
  long ed = gid >> 4;
  if (ed >= e) return;
  int f = (int)(gid & 15) * 4;
  int s = src[ed], d = dst[ed];
  float4 v = *reinterpret_cast<const float4*>(h + (size_t)s * D + f);
  float* p = neigh + (size_t)d * D + f;
  atomicAdd(p + 0, v.x);
  atomicAdd(p + 1, v.y);
  atomicAdd(p + 2, v.z);
  atomicAdd(p + 3, v.w);
}

// out = h @ Ws + (neigh/deg) @ Wn + bias ; optional relu + row l2norm
// one wave per 16-row tile; fp32 WMMA 16x16x4; weights staged transposed in LDS
__global__ void __launch_bounds__(256)
k_sage(const float* __restrict__ h, const float* __restrict__ neigh,
       const float* __restrict__ deg, const float* __restrict__ Ws,
       const float* __restrict__ Wn, const float* __restrict__ bias,
       float* __restrict__ out, int n, int do_norm) {
  __shared__ float wls[D][WPAD];
  __shared__ float wln[D][WPAD];
  stage_wT(Ws, wls, threadIdx.x);
  stage_wT(Wn, wln, threadIdx.x);
  __syncthreads();  // all waves participate (guard comes after)

  const int lane = threadIdx.x & 31;
  const int wave = threadIdx.x >> 5;
  const int row0 = (blockIdx.x * 8 + wave) * 16;
  if (row0 >= n) return;
  const int half = lane >> 4;
  const int l15 = lane & 15;
  int rowA = row0 + l15;
  if (rowA >= n) rowA = n - 1;  // tail-safe loads
  const float dinv = 1.0f / fmaxf(deg[rowA], 1.0f);
  const float* __restrict__ hp = h + (size_t)rowA * D;
  const float* __restrict__ np = neigh + (size_t)rowA * D;

  v8f acc[4] = {};
  for (int k0 = 0; k0 < D; k0 += 4) {
    const int kk = k0 + half * 2;
    const v2f ah = *reinterpret_cast<const v2f*>(hp + kk);
    v2f an = *reinterpret_cast<const v2f*>(np + kk);
    an.x *= dinv;
    an.y *= dinv;
#pragma unroll
    for (int t = 0; t < 4; ++t) {
      const int nn = t * 16 + l15;
      const v2f bs = *reinterpret_cast<const v2f*>(&wls[nn][kk]);  // ds_load_b64
      const v2f bn = *reinterpret_cast<const v2f*>(&wln[nn][kk]);
      acc[t] = WMMA4(ah, bs, acc[t]);
      acc[t] = WMMA4(an, bn, acc[t]);
    }
  }

  float ss[8];
#pragma unroll
  for (int i = 0; i < 8; ++i) ss[i] = 0.0f;
#pragma unroll
  for (int t = 0; t < 4; ++t) {
    const float bv = bias[t * 16 + l15];
#pragma unroll
    for (int i = 0; i < 8; ++i) {
      float v = acc[t][i] + bv;
      if (do_norm) v = fmaxf(v, 0.0f);
      acc[t][i] = v;
      ss[i] += v * v;
    }
  }
  if (do_norm) {
#pragma unroll
    for (int i = 0; i < 8; ++i) {
      float s = ss[i];
      s += __shfl_xor(s, 1, 32);
      s += __shfl_xor(s, 2, 32);
      s += __shfl_xor(s, 4, 32);
      s += __shfl_xor(s, 8, 32);  // reduction stays within each 16-lane half
      const float inv = 1.0f / fmaxf(sqrtf(s), 1e-12f);
#pragma unroll
      for (int t = 0; t < 4; ++t) acc[t][i] *= inv;
    }
  }
#pragma unroll
  for (int t = 0; t < 4; ++t) {
#pragma unroll
    for (int i = 0; i < 8; ++i) {
      const int r = row0 + i + half * 8;
      if (r < n) out[(size_t)r * D + t * 16 + l15] = acc[t][i];
    }
  }
}

// head: u = softplus(relu(BN(h@fcw + fcb))); z_loc = u@w21+b21; z_scale = exp(u@w22+b22)
__global__ void __launch_bounds__(256)
k_head(const float* __restrict__ h, const float* __restrict__ fcw,
       const float* __restrict__ fcb, const float* __restrict__ gamma,
       const float* __restrict__ beta, const float* __restrict__ mean,
       const float* __restrict__ var, const float* __restrict__ w21,
       const float* __restrict__ b21, const float* __restrict__ w22,
       const float* __restrict__ b22, float* __restrict__ zloc,
       float* __restrict__ zscale, int n) {
  __shared__ float wl0[D][WPAD];  // fcw^T
  __shared__ float wl1[D][WPAD];  // w21^T
  __shared__ float wl2[D][WPAD];  // w22^T
  __shared__ float u[8][16][WPAD];  // wave-private intermediate, padded (bank-safe)
  stage_wT(fcw, wl0, threadIdx.x);
  stage_wT(w21, wl1, threadIdx.x);
  stage_wT(w22, wl2, threadIdx.x);
  __syncthreads();

  const int lane = threadIdx.x & 31;
  const int wave = threadIdx.x >> 5;
  const int row0 = (blockIdx.x * 8 + wave) * 16;
  if (row0 >= n) return;
  const int half = lane >> 4;
  const int l15 = lane & 15;
  int rowA = row0 + l15;
  if (rowA >= n) rowA = n - 1;
  const float* __restrict__ hp = h + (size_t)rowA * D;

  // GEMM 1: h @ fcw
  v8f acc[4] = {};
  for (int k0 = 0; k0 < D; k0 += 4) {
    const int kk = k0 + half * 2;
    const v2f a = *reinterpret_cast<const v2f*>(hp + kk);
#pragma unroll
    for (int t = 0; t < 4; ++t) {
      const int nn = t * 16 + l15;
      const v2f b = *reinterpret_cast<const v2f*>(&wl0[nn][kk]);
      acc[t] = WMMA4(a, b, acc[t]);
    }
  }
  // epilogue: bias -> BN(eval) -> relu -> softplus ; park in LDS (C-layout -> row-major)
#pragma unroll
  for (int t = 0; t < 4; ++t) {
    const int c = t * 16 + l15;
    const float bb = fcb[c];
    const float m = mean[c];
    const float iv = rsqrtf(var[c] + 1e-5f);
    const float g = gamma[c];
    const float bt = beta[c];
#pragma unroll
    for (int i = 0; i < 8; ++i) {
      float v = acc[t][i] + bb;
      v = (v - m) * iv * g + bt;
      v = fmaxf(v, 0.0f);
      v = v + __logf(1.0f + __expf(-v));  // softplus, stable for v >= 0
      u[wave][i + half * 8][c] = v;
    }
  }
  // wave-private LDS RAW (compiler inserts s_wait_dscnt); no block barrier needed
  // GEMM 2 & 3 from LDS (A-layout: row = l15, k = k0 + 2*half)
  v8f aL[4] = {}, aS[4] = {};
  for (int k0 = 0; k0 < D; k0 += 4) {
    const int kk = k0 + half * 2;
    const v2f a = *reinterpret_cast<const v2f*>(&u[wave][l15][kk]);
#pragma unroll
    for (int t = 0; t < 4; ++t) {
      const int nn = t * 16 + l15;
      const v2f bL = *reinterpret_cast<const v2f*>(&wl1[nn][kk]);
      const v2f bS = *reinterpret_cast<const v2f*>(&wl2[nn][kk]);
      aL[t] = WMMA4(a, bL, aL[t]);
      aS[t] = WMMA4(a, bS, aS[t]);
    }
  }
#pragma unroll
  for (int t = 0; t < 4; ++t) {
    const int c = t * 16 + l15;
    const float bl = b21[c];
    const float bs = b22[c];
#pragma unroll
    for (int i = 0; i < 8; ++i) {
      const int r = row0 + i + half * 8;
      if (r < n) {
        const size_t o = (size_t)r * D + c;
        zloc[o] = aL[t][i] + bl;
        zscale[o] = __expf(aS[t][i] + bs);
      }
    }
  }
}

extern "C" void kernel_launch(void* const* d_in, const int* in_sizes, int n_in,
                              void* d_out, int out_size, void* d_ws, size_t ws_size,
                              hipStream_t stream) {
  const float* x = (const float*)d_in[0];
  const int* esrc = (const int*)d_in[1];
  const int* edst = (const int*)d_in[2];
  const float* Wself = (const float*)d_in[3];
  const float* Wneigh = (const float*)d_in[4];
  const float* bsage = (const float*)d_in[5];
  const float* fcw = (const float*)d_in[6];
  const float* fcb = (const float*)d_in[7];
  const float* gamma = (const float*)d_in[8];
  const float* beta = (const float*)d_in[9];
  const float* mean = (const float*)d_in[10];
  const float* var = (const float*)d_in[11];
  const float* w21 = (const float*)d_in[12];
  const float* b21 = (const float*)d_in[13];
  const float* w22 = (const float*)d_in[14];
  const float* b22 = (const float*)d_in[15];

  const int n = in_sizes[0] / D;
  const int e = in_sizes[1];
  const size_t nd = (size_t)n * D;

  // workspace layout: deg | hA | hB | neigh  (~77 MB for N=100K)
  float* ws = (float*)d_ws;
  float* deg = ws;
  float* hA = ws + (((size_t)n + 255) & ~(size_t)255);
  float* hB = hA + nd;
  float* nb = hB + nd;

  {
    long tot = (long)n * D;
    k_init<<<(int)((tot + 255) / 256), 256, 0, stream>>>(x, hA, deg, n);
  }
  k_deg<<<(e + 255) / 256, 256, 0, stream>>>(edst, deg, e);

  const int tiles = (n + 15) / 16;
  const int gblocks = (tiles + 7) / 8;

  float* hcur = hA;
  float* hnext = hB;
  for (int l = 0; l < 3; ++l) {
    long n4 = (long)(nd / 4);
    k_zero4<<<(int)((n4 + 255) / 256), 256, 0, stream>>>((float4*)nb, n4);
    long st = (long)e * 16;
    k_scatter<<<(int)((st + 255) / 256), 256, 0, stream>>>(hcur, esrc, edst, nb, e);
    k_sage<<<gblocks, 256, 0, stream>>>(hcur, nb, deg, Wself + (size_t)l * D * D,
                                        Wneigh + (size_t)l * D * D, bsage + (size_t)l * D,
                                        hnext, n, (l < 2) ? 1 : 0);
    float* tmp = hcur;
    hcur = hnext;
    hnext = tmp;
  }

  float* zloc = (float*)d_out;
  float* zscale = zloc + nd;
  k_head<<<gblocks, 256, 0, stream>>>(hcur, fcw, fcb, gamma, beta, mean, var, w21, b21,
                                      w22, b22, zloc, zscale, n);
}